// JaggedMaxModule_39762807226830
// MI455X (gfx1250) — compile-verified
//
#include <hip/hip_runtime.h>

// Jagged per-segment max on MI455X (gfx1250, wave32).
//
// values : [TOTAL, 32] f32   (TOTAL = 3.2M)
// prefix : [NSEG]      i32   sorted exclusive end offsets, in [0, TOTAL)
// out    : [NSEG, 32]  f32   segment s = max over rows [prefix[s-1], prefix[s]);
//                            empty segment -> -inf (segment_max identity)
//
// Pure streaming reduction: ~410 MB read once (> 192 MB L2) -> NT hints;
// roofline floor ~18 us @ 23.3 TB/s. One wave32 per segment. Main loop is a
// CDNA5 async pipeline: global_load_async_to_lds_b128 double-buffers 512 B
// chunks (4 rows) per wave, s_wait_asynccnt keeps exactly one chunk in
// flight while the previous one is reduced out of LDS via ds_load_b128.

typedef float v4f __attribute__((ext_vector_type(4)));

#define DIMF 32                 // floats per row
#define WAVES_PER_BLOCK 8
#define CHUNK_FLOATS 128        // 4 rows * 32 cols
#define BUF_FLOATS  (2 * CHUNK_FLOATS)   // double buffer per wave

__global__ __launch_bounds__(256)
void jagged_segment_max_kernel(const float* __restrict__ values,
                               const int*   __restrict__ prefix,
                               float*       __restrict__ out,
                               int nseg)
{
    // 8 waves * 2 buffers * 512 B = 8 KB LDS per block
    __shared__ __align__(16) float stage[WAVES_PER_BLOCK * BUF_FLOATS];

    const int wid   = (int)((blockIdx.x * blockDim.x + threadIdx.x) >> 5); // wave = segment
    const int lane  = (int)(threadIdx.x & 31);
    const int wslot = (int)(threadIdx.x >> 5);
    if (wid >= nseg) return;

    const int end   = prefix[wid];
    const int start = (wid == 0) ? 0 : prefix[wid - 1];

    const int rowsub = lane >> 3;   // 0..3 : row-phase inside a 4-row chunk
    const int col4   = lane & 7;    // 0..7 : which float4 of the 32-wide row

    const float NEG_INF = -__builtin_inff();
    v4f acc = { NEG_INF, NEG_INF, NEG_INF, NEG_INF };

    // This lane's fixed 16 B slice inside any 4-row chunk.
    const size_t lane_goff = (size_t)rowsub * DIMF + (size_t)col4 * 4;

    float* wbuf = &stage[wslot * BUF_FLOATS];
    // Wave-logical LDS byte address of this lane's slot in buffer 0
    // (flat->LDS mapping uses addr[31:0] as the logical LDS address).
    const unsigned lds0 = (unsigned)(size_t)wbuf + (unsigned)(lane * 16);

    const int nchunks = (end - start) >> 2;   // full 4-row chunks

    if (nchunks > 0) {
        const float* g0 = values + (size_t)start * DIMF + lane_goff;
        asm volatile("global_load_async_to_lds_b128 %0, %1, off"
                     :: "v"(lds0), "v"(g0) : "memory");
    }
    for (int i = 0; i < nchunks; ++i) {
        if (i + 1 < nchunks) {
            // Launch chunk i+1 into the other buffer, then wait for chunk i.
            const float* gn =
                values + (size_t)(start + ((i + 1) << 2)) * DIMF + lane_goff;
            const unsigned ldsn = lds0 + (((i + 1) & 1) ? 512u : 0u);
            asm volatile("global_load_async_to_lds_b128 %0, %1, off"
                         :: "v"(ldsn), "v"(gn) : "memory");
            asm volatile("s_wait_asynccnt 0x1" ::: "memory");
        } else {
            asm volatile("s_wait_asynccnt 0x0" ::: "memory");
        }
        const v4f* pb = reinterpret_cast<const v4f*>(
            &wbuf[((i & 1) ? CHUNK_FLOATS : 0) + lane * 4]);
        v4f v = *pb;                          // ds_load_b128
        acc.x = fmaxf(acc.x, v.x);
        acc.y = fmaxf(acc.y, v.y);
        acc.z = fmaxf(acc.z, v.z);
        acc.w = fmaxf(acc.w, v.w);
    }

    // Tail: 0..3 leftover rows, predicated direct NT loads.
    const int r = start + (nchunks << 2);
    if (rowsub < end - r) {
        const v4f* p =
            reinterpret_cast<const v4f*>(values + (size_t)(r + rowsub) * DIMF) + col4;
        v4f v = __builtin_nontemporal_load(p);  // global_load_b128, TH=NT
        acc.x = fmaxf(acc.x, v.x);
        acc.y = fmaxf(acc.y, v.y);
        acc.z = fmaxf(acc.z, v.z);
        acc.w = fmaxf(acc.w, v.w);
    }

    // Fold the 4 row-phases: combine lanes differing in bits 3 and 4 (wave32).
    #pragma unroll
    for (int mask = 8; mask <= 16; mask <<= 1) {
        acc.x = fmaxf(acc.x, __shfl_xor(acc.x, mask, 32));
        acc.y = fmaxf(acc.y, __shfl_xor(acc.y, mask, 32));
        acc.z = fmaxf(acc.z, __shfl_xor(acc.z, mask, 32));
        acc.w = fmaxf(acc.w, __shfl_xor(acc.w, mask, 32));
    }

    // Lanes 0..7 (rowsub==0, col4==lane) hold the 32 final columns: 128 B store.
    if (lane < 8) {
        v4f* q = reinterpret_cast<v4f*>(out + (size_t)wid * DIMF) + lane;
        __builtin_nontemporal_store(acc, q);    // written once, never re-read
    }
}

extern "C" void kernel_launch(void* const* d_in, const int* in_sizes, int n_in,
                              void* d_out, int out_size, void* d_ws, size_t ws_size,
                              hipStream_t stream)
{
    const float* values = (const float*)d_in[0];
    const int*   prefix = (const int*)d_in[1];   // integer inputs delivered as i32
    float*       out    = (float*)d_out;

    const int nseg   = in_sizes[1];              // 100,000
    const int blocks = (nseg + WAVES_PER_BLOCK - 1) / WAVES_PER_BLOCK;

    jagged_segment_max_kernel<<<dim3(blocks), dim3(WAVES_PER_BLOCK * 32), 0, stream>>>(
        values, prefix, out, nseg);
}